// ASPECTS_multiloss_80685255623232
// MI455X (gfx1250) — compile-verified
//
#include <hip/hip_runtime.h>
#include <math.h>

typedef float v2f __attribute__((ext_vector_type(2)));
typedef float v4f __attribute__((ext_vector_type(4)));
typedef float v8f __attribute__((ext_vector_type(8)));

#define TPB   256
#define NBLK  2048
#define WAVES (TPB / 32)

// focal term with alpha==1:  alpha_t = y; term = y * (1 - e^{-bce})^2 * bce
__device__ __forceinline__ float focal_term(float x, float y) {
    float t   = expf(-fabsf(x));                       // e^{-|x|}
    float bce = fmaxf(x, 0.0f) - x * y + log1pf(t);    // stable BCE-with-logits
    float pt  = expf(-bce);
    float om  = 1.0f - pt;
    return y * om * om * bce;
}

__global__ __launch_bounds__(TPB) void aspects_stage1(
    const float* __restrict__ x, const float* __restrict__ y,
    const float* __restrict__ hs_w, const float* __restrict__ hs_b,
    float* __restrict__ partial, int rows) {

    const float w = hs_w[0];     // uniform -> scalar loads
    const float b = hs_b[0];

    float s_focal = 0.0f, s_aspect = 0.0f, s_detect = 0.0f, s_cs = 0.0f;

    for (int r = blockIdx.x * TPB + threadIdx.x; r < rows; r += TPB * NBLK) {
        const v4f* xv = (const v4f*)(x + (size_t)r * 20);   // 80B rows, 16B aligned
        const v4f* yv = (const v4f*)(y + (size_t)r * 20);
        v4f xq[5], yq[5];
#pragma unroll
        for (int i = 0; i < 5; ++i) {
            xq[i] = __builtin_nontemporal_load(xv + i);     // streaming: NT hint
            yq[i] = __builtin_nontemporal_load(yv + i);
        }

        float ysum[2] = {0.0f, 0.0f};
        float xsum[2] = {0.0f, 0.0f};
        float xmin[2] = {3.402823466e38f, 3.402823466e38f};

#pragma unroll
        for (int k = 0; k < 20; ++k) {
            float xe = xq[k >> 2][k & 3];
            float ye = yq[k >> 2][k & 3];
            s_focal += focal_term(xe, ye);                  // main focal (mean over 20B)
            int c = k & 1;
            ysum[c] += ye;
            xsum[c] += xe;
            xmin[c] = fminf(xmin[c], xe);
        }

#pragma unroll
        for (int c = 0; c < 2; ++c) {
            float yt = (ysum[c] >= 6.0f)  ? 1.0f : 0.0f;    // ASPECT_TH
            float xh = xsum[c] * 0.1f * w + b;              // mean over 10, 1x1 conv
            s_aspect += focal_term(xh, yt);
            float yd = (ysum[c] >= 10.0f) ? 1.0f : 0.0f;    // DETECT_TH
            s_detect += focal_term(xmin[c], yd);
        }

#pragma unroll
        for (int k = 0; k < 20; ++k) {
            float xe = xq[k >> 2][k & 3];
            s_cs += fmaxf(-xe, 0.0f) * fmaxf(xmin[k & 1], 0.0f);
        }
    }

    // wave32 shuffle reduce + LDS across 8 waves
    __shared__ float red[4][WAVES];
    float vals[4] = {s_focal, s_aspect, s_detect, s_cs};
#pragma unroll
    for (int a = 0; a < 4; ++a) {
        float v = vals[a];
#pragma unroll
        for (int off = 16; off > 0; off >>= 1) v += __shfl_down(v, off, 32);
        if ((threadIdx.x & 31) == 0) red[a][threadIdx.x >> 5] = v;
    }
    __syncthreads();
    if (threadIdx.x < 4) {
        float v = 0.0f;
#pragma unroll
        for (int wv = 0; wv < WAVES; ++wv) v += red[threadIdx.x][wv];
        partial[threadIdx.x * NBLK + blockIdx.x] = v;   // accumulator-major [4][NBLK]
    }
}

// Stage 2: one wave. WMMA f32 16x16x4 as a 64-way accumulating adder:
// D = A(data) x B(ones) + C  =>  D[m,n] = sum_k A[m,k] + C[m,n].
// After NBLK/64 iterations, sum of column 0 (lane0: M0..7 in c[0..7],
// lane16: M8..15) equals the exact sum of all NBLK values.
__global__ __launch_bounds__(32) void aspects_stage2(
    const float* __restrict__ partial, float* __restrict__ out,
    float inv20B, float inv2B) {

    int lane = threadIdx.x;
    v2f ones; ones[0] = 1.0f; ones[1] = 1.0f;

    float totals[4];
#pragma unroll
    for (int a = 0; a < 4; ++a) {
        const float* p = partial + a * NBLK;
        v8f c = {};
        for (int base = 0; base < NBLK; base += 64) {
            v2f am;
            am[0] = p[base + 2 * lane];
            am[1] = p[base + 2 * lane + 1];
            c = __builtin_amdgcn_wmma_f32_16x16x4_f32(
                    /*neg_a=*/false, am, /*neg_b=*/false, ones,
                    /*c_mod=*/(short)0, c, /*reuse_a=*/false, /*reuse_b=*/false);
        }
        float local = c[0] + c[1] + c[2] + c[3] + c[4] + c[5] + c[6] + c[7];
        totals[a] = __shfl(local, 0, 32) + __shfl(local, 16, 32);
    }

    if (lane == 0) {
        // focal/(20B) + 1.0*aspect/(2B) + 0.5*cs/(20B) + 0.5*detect/(2B)
        out[0] = totals[0] * inv20B
               + totals[1] * inv2B
               + 0.5f * totals[3] * inv20B
               + 0.5f * totals[2] * inv2B;
    }
}

extern "C" void kernel_launch(void* const* d_in, const int* in_sizes, int n_in,
                              void* d_out, int out_size, void* d_ws, size_t ws_size,
                              hipStream_t stream) {
    const float* x    = (const float*)d_in[0];
    const float* y    = (const float*)d_in[1];
    const float* hs_w = (const float*)d_in[2];
    const float* hs_b = (const float*)d_in[3];
    int rows = in_sizes[0] / 20;

    float* partial = (float*)d_ws;   // 4 * NBLK floats = 32 KB

    aspects_stage1<<<NBLK, TPB, 0, stream>>>(x, y, hs_w, hs_b, partial, rows);

    float inv20B = 1.0f / (20.0f * (float)rows);
    float inv2B  = 1.0f / (2.0f  * (float)rows);
    aspects_stage2<<<1, 32, 0, stream>>>(partial, (float*)d_out, inv20B, inv2B);
}